// MLPPredictor_21251498181091
// MI455X (gfx1250) — compile-verified
//
#include <hip/hip_runtime.h>
#include <hip/hip_bf16.h>

// Edge-MLP (GNN edge predictor) for MI455X / gfx1250, wave32 + WMMA.
//
// Roofline: 59.6 GFLOP fp32; h (51MB) + weights live in the 192MB L2, so HBM
// traffic is ~100MB (~4us @ 23.3TB/s) -> matrix-pipe bound. We use
// V_WMMA_F32_16X16X4_F32 (fp32 in / fp32 accumulate) to match the fp32
// reference numerics while running on the matrix pipe.
//
// One wave32 owns a 16-edge tile. A-fragments (16x4 f32) come straight from
// the L2-resident gather of h; B-fragments come straight from the row-major
// [out,in] weight matrices (which is exactly the B-fragment access pattern).
// Inter-layer transpose (D layout -> A layout) goes through per-wave LDS
// tiles with bank-conflict-free padded strides; same-wave DS ordering makes
// s_wait_dscnt sufficient (no workgroup barrier -> waves stay independent).

typedef __attribute__((ext_vector_type(2))) float v2f;
typedef __attribute__((ext_vector_type(8))) float v8f;

#define HF   128          // node feature dim
#define KIN  256          // 2*HF, layer-1 K
#define S1   68           // LDS row stride (floats) for [16][64] act1 (68%64=4 -> conflict-free)
#define S2   36           // LDS row stride (floats) for [16][32] act2 (36%64=36 -> conflict-free)

static __device__ __forceinline__ v8f wmma_f32(v2f a, v2f b, v8f c) {
    // D = A(16x4 f32) x B(4x16 f32) + C(16x16 f32)
    return __builtin_amdgcn_wmma_f32_16x16x4_f32(
        /*neg_a=*/false, a, /*neg_b=*/false, b,
        /*c_mod=*/(short)0, c, /*reuse_a=*/false, /*reuse_b=*/false);
}

__global__ __launch_bounds__(256) void edge_mlp_wmma(
    const float* __restrict__ h,
    const int*   __restrict__ src,
    const int*   __restrict__ dst,
    const float* __restrict__ W1, const float* __restrict__ b1,
    const float* __restrict__ W2, const float* __restrict__ b2,
    const float* __restrict__ W3, const float* __restrict__ b3,
    float* __restrict__ out, int nEdges)
{
    __shared__ float lds1[8][16 * S1];   // per-wave layer-1 activations [16][64]
    __shared__ float lds2[8][16 * S2];   // per-wave layer-2 activations [16][32]

    const int lane   = threadIdx.x & 31;
    const int wave   = threadIdx.x >> 5;
    const int nTiles = (nEdges + 15) >> 4;
    const int tile   = blockIdx.x * 8 + wave;
    if (tile >= nTiles) return;          // wave-uniform: EXEC stays all-1s below

    const int e0   = tile << 4;
    const int m    = lane & 15;          // edge-row (for A) / N column (for B,C,D)
    const int kh   = lane >> 4;          // half-select: K pair 0..1 vs 2..3
    const int koff = kh << 1;

    // ---- per-lane gather pointers (clamped; stores are predicated later) ----
    int e = e0 + m;
    if (e >= nEdges) e = nEdges - 1;
    const float* pS = h + (size_t)src[e] * HF;
    const float* pD = h + (size_t)dst[e] * HF;

    // ================= Layer 1: X[16,256] @ W1^T -> [16,64] =================
    const float* pB0 = W1 + (size_t)( 0 + m) * KIN + koff;
    const float* pB1 = W1 + (size_t)(16 + m) * KIN + koff;
    const float* pB2 = W1 + (size_t)(32 + m) * KIN + koff;
    const float* pB3 = W1 + (size_t)(48 + m) * KIN + koff;

    v8f acc0 = {}, acc1 = {}, acc2 = {}, acc3 = {};

#pragma unroll 8
    for (int k = 0; k < HF; k += 4) {            // K = 0..127  (h[src] half)
        v2f a  = *(const v2f*)(pS + (k + koff));
        v2f q0 = *(const v2f*)(pB0 + k);
        v2f q1 = *(const v2f*)(pB1 + k);
        v2f q2 = *(const v2f*)(pB2 + k);
        v2f q3 = *(const v2f*)(pB3 + k);
        acc0 = wmma_f32(a, q0, acc0);
        acc1 = wmma_f32(a, q1, acc1);
        acc2 = wmma_f32(a, q2, acc2);
        acc3 = wmma_f32(a, q3, acc3);
    }
#pragma unroll 8
    for (int k = 0; k < HF; k += 4) {            // K = 128..255 (h[dst] half)
        v2f a  = *(const v2f*)(pD + (k + koff));
        v2f q0 = *(const v2f*)(pB0 + HF + k);
        v2f q1 = *(const v2f*)(pB1 + HF + k);
        v2f q2 = *(const v2f*)(pB2 + HF + k);
        v2f q3 = *(const v2f*)(pB3 + HF + k);
        acc0 = wmma_f32(a, q0, acc0);
        acc1 = wmma_f32(a, q1, acc1);
        acc2 = wmma_f32(a, q2, acc2);
        acc3 = wmma_f32(a, q3, acc3);
    }

    // bias + ReLU, scatter D layout -> row-major LDS tile (A layout source)
    float* X1 = lds1[wave];
    {
        float bv0 = b1[m], bv1 = b1[16 + m], bv2 = b1[32 + m], bv3 = b1[48 + m];
#pragma unroll
        for (int r = 0; r < 8; ++r) {
            int row = r + (kh << 3);
            float t0 = acc0[r] + bv0; X1[row * S1 +      m] = t0 > 0.f ? t0 : 0.f;
            float t1 = acc1[r] + bv1; X1[row * S1 + 16 + m] = t1 > 0.f ? t1 : 0.f;
            float t2 = acc2[r] + bv2; X1[row * S1 + 32 + m] = t2 > 0.f ? t2 : 0.f;
            float t3 = acc3[r] + bv3; X1[row * S1 + 48 + m] = t3 > 0.f ? t3 : 0.f;
        }
    }
    // same-wave DS ops are in-order; wait + memory clobber for cross-lane RAW
    asm volatile("s_wait_dscnt 0" ::: "memory");

    // ================= Layer 2: X1[16,64] @ W2^T -> [16,32] =================
    const float* pC0 = W2 + (size_t)( 0 + m) * 64 + koff;
    const float* pC1 = W2 + (size_t)(16 + m) * 64 + koff;
    v8f c20 = {}, c21 = {};
#pragma unroll
    for (int k = 0; k < 64; k += 4) {
        v2f a  = *(const v2f*)(X1 + m * S1 + k + koff);   // ds_load_b64
        v2f q0 = *(const v2f*)(pC0 + k);
        v2f q1 = *(const v2f*)(pC1 + k);
        c20 = wmma_f32(a, q0, c20);
        c21 = wmma_f32(a, q1, c21);
    }

    float* X2 = lds2[wave];
    {
        float bv0 = b2[m], bv1 = b2[16 + m];
#pragma unroll
        for (int r = 0; r < 8; ++r) {
            int row = r + (kh << 3);
            float t0 = c20[r] + bv0; X2[row * S2 +      m] = t0 > 0.f ? t0 : 0.f;
            float t1 = c21[r] + bv1; X2[row * S2 + 16 + m] = t1 > 0.f ? t1 : 0.f;
        }
    }
    asm volatile("s_wait_dscnt 0" ::: "memory");

    // ================= Layer 3: X2[16,32] @ W3^T -> [16,6] ==================
    // Only 6 valid N columns: clamp row + multiplicative mask so EXEC stays
    // all-1s through the WMMAs (no divergent branch before matrix ops).
    const int   wr  = (m < 6) ? m : 5;
    const float msk = (m < 6) ? 1.f : 0.f;
    const float* pC3 = W3 + (size_t)wr * 32 + koff;
    v8f c3 = {};
#pragma unroll
    for (int k = 0; k < 32; k += 4) {
        v2f a = *(const v2f*)(X2 + m * S2 + k + koff);
        v2f q = *(const v2f*)(pC3 + k);
        q.x *= msk; q.y *= msk;
        c3 = wmma_f32(a, q, c3);
    }

    // ---- epilogue: bias + predicated store of [16,6] tile ----
    if (m < 6) {
        float bv = b3[m];
#pragma unroll
        for (int r = 0; r < 8; ++r) {
            int eo = e0 + r + (kh << 3);
            if (eo < nEdges) out[(size_t)eo * 6 + m] = c3[r] + bv;
        }
    }
}

extern "C" void kernel_launch(void* const* d_in, const int* in_sizes, int n_in,
                              void* d_out, int out_size, void* d_ws, size_t ws_size,
                              hipStream_t stream) {
    (void)n_in; (void)out_size; (void)d_ws; (void)ws_size;
    const float* h   = (const float*)d_in[0];
    const int*   src = (const int*)  d_in[1];
    const int*   dst = (const int*)  d_in[2];
    const float* W1  = (const float*)d_in[3];
    const float* b1  = (const float*)d_in[4];
    const float* W2  = (const float*)d_in[5];
    const float* b2  = (const float*)d_in[6];
    const float* W3  = (const float*)d_in[7];
    const float* b3  = (const float*)d_in[8];
    float* out = (float*)d_out;

    const int nEdges = in_sizes[1];              // src element count == E
    const int nTiles = (nEdges + 15) / 16;       // 16 edges per wave
    const int blocks = (nTiles + 7) / 8;         // 8 waves per 256-thread block

    edge_mlp_wmma<<<blocks, 256, 0, stream>>>(h, src, dst, W1, b1, W2, b2,
                                              W3, b3, out, nEdges);
}